// BoundaryLoss_75170517615226
// MI455X (gfx1250) — compile-verified
//
#include <hip/hip_runtime.h>
#include <math.h>

// Problem geometry from the reference: (B=8, C=1, H=256, W=256)
#define HDIM 256
#define WDIM 256
#define BIGF 1.0e10f
#define RED_BLOCKS 64
#define WAVES_PER_BLOCK 8   // 256 threads / wave32

typedef __attribute__((ext_vector_type(2))) float v2f;
typedef __attribute__((ext_vector_type(8))) float v8f;

// gfx1250 async direct-to-LDS path (ASYNCcnt) -- guarded so either toolchain compiles
#if __has_builtin(__builtin_amdgcn_global_load_async_to_lds_b32) && \
    __has_builtin(__builtin_amdgcn_s_wait_asynccnt)
#define HAVE_ASYNC_LDS 1
#else
#define HAVE_ASYNC_LDS 0
#endif

__device__ __forceinline__ float sigmoidf_dev(float x) {
  return 1.0f / (1.0f + expf(-x));
}

#if HAVE_ASYNC_LDS
// Signature (from compiler diagnostic): first param is global ('__device__') int*,
// so pass int-typed addrspace(1)/(3) pointers: (gsrc, lds_dst, imm offset, cpol).
__device__ __forceinline__ void async_copy_b32(const float* gsrc, float* ldst) {
  __builtin_amdgcn_global_load_async_to_lds_b32(
      (__attribute__((address_space(1))) int*)(void*)const_cast<float*>(gsrc),
      (__attribute__((address_space(3))) int*)(void*)ldst,
      /*imm offset=*/0, /*cpol=*/0);
}
#endif

// ---------------------------------------------------------------------------
// K0: reset the global max-abs atomic slot (uint bits of a nonnegative float)
// ---------------------------------------------------------------------------
__global__ void k_init(unsigned int* maxslot) { *maxslot = 0u; }

// ---------------------------------------------------------------------------
// K1: vertical 1-D pass per column. One thread per (image, column).
// Forward scan stores tentative "distance to nearest feature above" (float);
// backward scan merges with "distance below", squares and caps -> g2.
// Integer distances <= 255 are exact in f32; empty-column sentinel -> 1e10f,
// matching the reference (1e10f +- row rounds back to 1e10f, g*g capped).
// Writes are coalesced: at step i, threads j=0..255 write one contiguous row.
// ---------------------------------------------------------------------------
__global__ void k_vert(const int* __restrict__ target,
                       float* __restrict__ g2p,
                       float* __restrict__ g2n) {
  const int b = blockIdx.x;       // image index (B*C)
  const int j = threadIdx.x;      // column
  const size_t base = (size_t)b * HDIM * WDIM;
  const int* t = target + base;
  float* gp = g2p + base;
  float* gn = g2n + base;

  const int NONE = 1 << 20;       // "no feature yet" sentinel distance
  int lastP = -NONE, lastN = -NONE;
  for (int i = 0; i < HDIM; ++i) {
    const size_t idx = (size_t)i * WDIM + j;
    const int m = (t[idx] > 0);
    if (m) lastP = i; else lastN = i;
    gp[idx] = (float)(i - lastP);     // distance to nearest m==1 above
    gn[idx] = (float)(i - lastN);     // distance to nearest m==0 above
  }

  int nextP = NONE, nextN = NONE;
  for (int i = HDIM - 1; i >= 0; --i) {
    const size_t idx = (size_t)i * WDIM + j;
    const int m = (t[idx] > 0);
    if (m) nextP = i; else nextN = i;
    const float gP = fminf(gp[idx], (float)(nextP - i));
    const float gN = fminf(gn[idx], (float)(nextN - i));
    gp[idx] = (gP < (float)HDIM) ? gP * gP : BIGF;   // min(g*g, BIG)
    gn[idx] = (gN < (float)HDIM) ? gN * gN : BIGF;
  }
}

// ---------------------------------------------------------------------------
// K2: lower envelope along rows (exactly the reference's brute-force min):
//     D2[i,j] = min_k g2[k,j] + (i-k)^2
// One block per (column j, image b); thread i handles one output row.
// Column of g2 staged in LDS -- via the gfx1250 async direct-to-LDS path
// (GLOBAL_LOAD_ASYNC_TO_LDS_B32 + s_wait_asynccnt) when available, skipping
// the VGPR round-trip; VGPR staging otherwise. Then signed distance, block
// max-abs reduction, deterministic (order-independent) uint atomicMax.
// ---------------------------------------------------------------------------
__global__ void k_env(const float* __restrict__ g2p,
                      const float* __restrict__ g2n,
                      float* __restrict__ dmap,
                      unsigned int* __restrict__ maxslot) {
  __shared__ float sp[HDIM];
  __shared__ float sn[HDIM];
  __shared__ float red[HDIM];

  const int j = blockIdx.x;
  const int b = blockIdx.y;
  const int i = threadIdx.x;
  const size_t base = (size_t)b * HDIM * WDIM;
  const size_t idx = base + (size_t)i * WDIM + j;

#if HAVE_ASYNC_LDS
  async_copy_b32(&g2p[idx], &sp[i]);     // ASYNCcnt-tracked, each lane copies one float from global memory into sp[i] in LDS
  async_copy_b32(&g2n[idx], &sn[i]);
  __builtin_amdgcn_s_wait_asynccnt(0);   // drain this wave's async copies
#else
  sp[i] = g2p[idx];
  sn[i] = g2n[idx];
#endif
  __syncthreads();

  float dp = 3.0e38f, dn = 3.0e38f;
  #pragma unroll 4
  for (int k = 0; k < HDIM; ++k) {
    const float ik = (float)(i - k);
    const float q = ik * ik;
    dp = fminf(dp, sp[k] + q);
    dn = fminf(dn, sn[k] + q);
  }

  const float dval = sqrtf(dp) - sqrtf(dn);
  dmap[idx] = dval;

  red[i] = fabsf(dval);
  __syncthreads();
  for (int s = HDIM / 2; s > 0; s >>= 1) {
    if (i < s) red[i] = fmaxf(red[i], red[i + s]);
    __syncthreads();
  }
  if (i == 0) atomicMax(maxslot, __float_as_uint(red[0]));  // nonneg: bit order == float order
}

// ---------------------------------------------------------------------------
// K3: Sum_e sigmoid(pred[e]) * dmap[e] using V_WMMA_F32_16X16X4_F32.
// B-matrix = all ones makes the layout irrelevant for a total sum:
//   D[m][n] += sum_k A[m][k], every column of D identical.
// Each wave packs 64 products per iteration into A (2 f32 VGPRs/lane) and
// accumulates into a 16x16 f32 C. Two independent accumulators break the
// D->C chain so back-to-back v_wmma issues can overlap in the XDL pipe.
// Full EXEC, wave-uniform loop bounds -> satisfies WMMA EXEC restrictions.
// ---------------------------------------------------------------------------
__global__ void k_reduce(const float* __restrict__ pred,
                         const float* __restrict__ dmap,
                         float* __restrict__ partials,
                         int N) {
  __shared__ float wpart[WAVES_PER_BLOCK];

  const int lane = threadIdx.x & 31;
  const int wib  = threadIdx.x >> 5;                 // wave in block
  const int wv   = blockIdx.x * WAVES_PER_BLOCK + wib;
  const int totalWaves = gridDim.x * WAVES_PER_BLOCK;
  const int nchunks = N >> 6;                        // 64 elements per chunk

  float wsum = 0.0f;

#if __has_builtin(__builtin_amdgcn_wmma_f32_16x16x4_f32)
  v8f acc0 = {};
  v8f acc1 = {};
  v2f bones;
  bones[0] = 1.0f;
  bones[1] = 1.0f;

  int c = wv;
  const int stride = totalWaves;
  // dual-accumulator main loop: chunks c -> acc0, c+stride -> acc1
  for (; c + stride < nchunks; c += 2 * stride) {
    const int f0 = c * 64 + lane;
    const int f1 = (c + stride) * 64 + lane;
    const int cn = c + 2 * stride;
    if (cn < nchunks) {                               // gfx1250 global_prefetch_b8
      __builtin_prefetch(&pred[cn * 64 + lane], 0, 1);
      __builtin_prefetch(&dmap[cn * 64 + lane], 0, 1);
    }
    v2f a0, a1;
    a0[0] = sigmoidf_dev(pred[f0])      * dmap[f0];
    a0[1] = sigmoidf_dev(pred[f0 + 32]) * dmap[f0 + 32];
    a1[0] = sigmoidf_dev(pred[f1])      * dmap[f1];
    a1[1] = sigmoidf_dev(pred[f1 + 32]) * dmap[f1 + 32];
    acc0 = __builtin_amdgcn_wmma_f32_16x16x4_f32(
        false, a0, false, bones, (short)0, acc0, false, false);
    acc1 = __builtin_amdgcn_wmma_f32_16x16x4_f32(
        false, a1, false, bones, (short)0, acc1, false, false);
  }
  // leftover chunk (at most one per wave)
  for (; c < nchunks; c += stride) {
    const int e0 = c * 64 + lane;
    v2f a;
    a[0] = sigmoidf_dev(pred[e0])      * dmap[e0];
    a[1] = sigmoidf_dev(pred[e0 + 32]) * dmap[e0 + 32];
    acc0 = __builtin_amdgcn_wmma_f32_16x16x4_f32(
        false, a, false, bones, (short)0, acc0, false, false);
  }
  acc0 = acc0 + acc1;                                 // merge accumulators
  float local = acc0[0] + acc0[1] + acc0[2] + acc0[3] +
                acc0[4] + acc0[5] + acc0[6] + acc0[7];
  // lane 0 holds rows M=0..7 of column N=0; lane 16 holds rows M=8..15.
  wsum = __shfl(local, 0, 32) + __shfl(local, 16, 32);
#else
  for (int c = wv; c < nchunks; c += totalWaves) {
    const int e0 = c * 64 + lane;
    wsum += sigmoidf_dev(pred[e0])      * dmap[e0];
    wsum += sigmoidf_dev(pred[e0 + 32]) * dmap[e0 + 32];
  }
  for (int off = 16; off > 0; off >>= 1) wsum += __shfl_xor(wsum, off, 32);
#endif

  // Remainder elements (none for N % 64 == 0, kept for generality)
  float sc = 0.0f;
  const int tid = blockIdx.x * blockDim.x + threadIdx.x;
  const int tot = gridDim.x * blockDim.x;
  for (int e = nchunks * 64 + tid; e < N; e += tot)
    sc += sigmoidf_dev(pred[e]) * dmap[e];
  for (int off = 16; off > 0; off >>= 1) sc += __shfl_xor(sc, off, 32);

  if (lane == 0) wpart[wib] = wsum + sc;
  __syncthreads();
  if (threadIdx.x == 0) {
    float s = 0.0f;
    for (int w = 0; w < WAVES_PER_BLOCK; ++w) s += wpart[w];
    partials[blockIdx.x] = s;
  }
}

// ---------------------------------------------------------------------------
// K4: deterministic finalize: loss = S / (N * max_abs)  (max_abs>0 else /N)
// ---------------------------------------------------------------------------
__global__ void k_final(const float* __restrict__ partials, int nparts,
                        const unsigned int* __restrict__ maxslot,
                        float* __restrict__ out, int N) {
  float s = 0.0f;
  for (int i = 0; i < nparts; ++i) s += partials[i];
  const float ma = __uint_as_float(*maxslot);
  const float denom = (ma > 0.0f) ? ma : 1.0f;
  out[0] = s / ((float)N * denom);
}

// ---------------------------------------------------------------------------
extern "C" void kernel_launch(void* const* d_in, const int* in_sizes, int n_in,
                              void* d_out, int out_size, void* d_ws, size_t ws_size,
                              hipStream_t stream) {
  (void)n_in; (void)out_size; (void)ws_size;
  const float* pred   = (const float*)d_in[0];
  const int*   target = (const int*)d_in[1];
  float* out = (float*)d_out;

  const int N  = in_sizes[0];                 // B*C*H*W = 524288
  const int BC = N / (HDIM * WDIM);           // 8 images

  // Workspace layout: g2_pos | g2_neg | dmap | partials | maxslot
  float* g2p = (float*)d_ws;
  float* g2n = g2p + N;
  float* dmap = g2n + N;
  float* partials = dmap + N;
  unsigned int* maxslot = (unsigned int*)(partials + RED_BLOCKS);

  k_init<<<1, 1, 0, stream>>>(maxslot);
  k_vert<<<BC, WDIM, 0, stream>>>(target, g2p, g2n);
  k_env<<<dim3(WDIM, BC), HDIM, 0, stream>>>(g2p, g2n, dmap, maxslot);
  k_reduce<<<RED_BLOCKS, 256, 0, stream>>>(pred, dmap, partials, N);
  k_final<<<1, 1, 0, stream>>>(partials, RED_BLOCKS, maxslot, out, N);
}